// GravityField_39462159515776
// MI455X (gfx1250) — compile-verified
//
#include <hip/hip_runtime.h>
#include <stdint.h>

// ---------------- problem constants (match setup_inputs) ----------------
#define N_     16
#define C_     64
#define W_     128
#define H_     128
#define WH_    (W_ * H_)          // 16384
#define WO_    32
#define HO_    32
#define CELLS_ (WO_ * HO_)        // 1024

// ---------------- workspace layout (floats) ----------------
#define M_OFF    0
#define DEN_OFF  (N_ * CELLS_)
#define CNT_OFF  (2 * N_ * CELLS_)
#define NRM_OFF  (3 * N_ * CELLS_)
#define CELL_OFF (NRM_OFF + N_ * WH_)
// total ws floats = 49152 + 2*262144 = 573440  (~2.3 MB)

// ---------------- gfx1250 async load-to-LDS probing ----------------
#ifndef __has_builtin
#define __has_builtin(x) 0
#endif

#define AS1 __attribute__((address_space(1)))
#define AS3 __attribute__((address_space(3)))

#if __has_builtin(__builtin_amdgcn_global_load_async_to_lds_b32)
#define HAVE_ASYNC_LDS 1
#else
#define HAVE_ASYNC_LDS 0
#endif

#if __has_builtin(__builtin_amdgcn_global_load_async_to_lds_b64)
#define HAVE_ASYNC_B64 1
#else
#define HAVE_ASYNC_B64 0
#endif

typedef int v2i_t __attribute__((ext_vector_type(2)));

// LDS row stride (floats) for the transposed X tile:
//  - b64 staging needs 8B-aligned rows -> 34 (34*l mod 64 distinct for l<32)
//  - b32 staging uses 33 (odd -> trivially conflict-free)
#if HAVE_ASYNC_B64
#define XSTR 34
#else
#define XSTR 33
#endif

__device__ __forceinline__ void wait_asynccnt0() {
#if __has_builtin(__builtin_amdgcn_s_wait_asynccnt)
  __builtin_amdgcn_s_wait_asynccnt(0);
#else
  asm volatile("s_wait_asynccnt 0" ::: "memory");
#endif
}

__device__ __forceinline__ void wait_dscnt0() {
  asm volatile("s_wait_dscnt 0" ::: "memory");
}

// ---------------- kernel A: zero ----------------
__global__ void k_zero(float* __restrict__ p, int n) {
  int i = blockIdx.x * blockDim.x + threadIdx.x;
  if (i < n) p[i] = 0.0f;
}

// ---------------- wave -> pixel-strip mapping ----------------
// gwave in [0,2048). Each wave owns a 128-pixel strip inside one quadrant of
// one image: w in {bw,bw+1}, h in [bh,bh+64). Cell is (near-)uniform per strip.
__device__ __forceinline__ void strip_base(int gwave, int& n, int& bw, int& bh) {
  n = gwave >> 7;                 // 128 strips per image
  int sid = gwave & 127;
  int qid = sid >> 5;             // quadrant 0..3
  int strip = sid & 31;
  int qx = qid & 1, qy = qid >> 1;
  bw = qx * 64 + strip * 2;
  bh = qy * 64;
}

__device__ __forceinline__ int cell_of(int w, int h, float omw) {
  float fx = (float)w * (1.0f / 64.0f) - 1.0f;
  float fy = (float)h * (1.0f / 64.0f) - 1.0f;
  float nl = sqrtf(fx * fx + fy * fy);
  nl = (nl < 1e-8f) ? 1.0f : nl;
  float inl = omw / nl;
  int zx = (int)rintf((fx * inl + 1.0f) * 15.5f);   // RNE == jnp.round
  int zy = (int)rintf((fy * inl + 1.0f) * 15.5f);
  zx = min(max(zx, 0), WO_ - 1);
  zy = min(max(zy, 0), HO_ - 1);
  return zx * HO_ + zy;
}

// ---------------- kernel B: norm, cell, scatter-max m ----------------
// One wave handles a whole 128-px strip: 4 consecutive pixels per lane,
// loaded as float4 (global_load_b128), results stored as float4/int4.
__global__ __launch_bounds__(128) void k_norm(const float* __restrict__ X,
                                              float* __restrict__ ws) {
  const int lane  = threadIdx.x & 31;
  const int gwave = (blockIdx.x * blockDim.x + threadIdx.x) >> 5;
  float* m    = ws + M_OFF;
  float* nrmC = ws + NRM_OFF;
  int*   celC = (int*)(ws + CELL_OFF);

  int n, bw, bh;
  strip_base(gwave, n, bw, bh);
  const int w  = bw + (lane >> 4);
  const int h0 = bh + 4 * (lane & 15);
  const int p0 = w * H_ + h0;                  // 16B-aligned pixel group
  const float* Xn = X + (size_t)n * C_ * WH_;

  float4 s = make_float4(0.f, 0.f, 0.f, 0.f);
#pragma unroll 8
  for (int c = 0; c < C_; ++c) {
    float4 x = *(const float4*)(Xn + c * WH_ + p0);
    s.x = fmaf(x.x, x.x, s.x);
    s.y = fmaf(x.y, x.y, s.y);
    s.z = fmaf(x.z, x.z, s.z);
    s.w = fmaf(x.w, x.w, s.w);
  }

  float nrm[4] = {sqrtf(s.x), sqrtf(s.y), sqrtf(s.z), sqrtf(s.w)};
  int cell[4];
#pragma unroll
  for (int j = 0; j < 4; ++j) {
    float omw = 1.0f - tanhf(nrm[j]);          // matches reference f32 (1 - w)
    cell[j] = cell_of(w, h0 + j, omw);
  }

  int gp0 = n * WH_ + p0;
  *(float4*)(nrmC + gp0) = make_float4(nrm[0], nrm[1], nrm[2], nrm[3]);
  *(int4*)(celC + gp0)   = make_int4(cell[0], cell[1], cell[2], cell[3]);

  int c0 = __builtin_amdgcn_readfirstlane(cell[0]);
  bool me = (cell[0] == c0) & (cell[1] == c0) & (cell[2] == c0) & (cell[3] == c0);
  if (__all(me)) {
    float mx = fmaxf(fmaxf(nrm[0], nrm[1]), fmaxf(nrm[2], nrm[3]));
#pragma unroll
    for (int o = 16; o > 0; o >>= 1) mx = fmaxf(mx, __shfl_xor(mx, o, 32));
    if (lane == 0)
      atomicMax((int*)&m[n * CELLS_ + c0], __float_as_int(mx));   // vals >= 0
  } else {
#pragma unroll
    for (int j = 0; j < 4; ++j)
      atomicMax((int*)&m[n * CELLS_ + cell[j]], __float_as_int(nrm[j]));
  }
}

// ---------------- kernel C: softmax-weighted scatter ----------------
#define WPB  4                   // waves per block (128 threads)
#define SLOT (C_ * XSTR + 32)    // per-wave LDS slot: X tile + e vector

__global__ __launch_bounds__(128) void k_scatter(const float* __restrict__ X,
                                                 float* __restrict__ ws,
                                                 float* __restrict__ num) {
  __shared__ float smem[WPB * SLOT];   // ~34.5 KB
  const int lane  = threadIdx.x & 31;
  const int wslot = threadIdx.x >> 5;
  const int gwave = (blockIdx.x * blockDim.x + threadIdx.x) >> 5;

  float* ldsX = &smem[wslot * SLOT];   // [C_][XSTR]
  float* ldsE = ldsX + C_ * XSTR;      // [32]

  const float* m    = ws + M_OFF;
  float*       den  = ws + DEN_OFF;
  float*       cnt  = ws + CNT_OFF;
  const float* nrmC = ws + NRM_OFF;
  const int*   celC = (const int*)(ws + CELL_OFF);

  int n, bw, bh;
  strip_base(gwave, n, bw, bh);
  const float* Xn = X + (size_t)n * C_ * WH_;

  const int cA = lane, cB = lane + 32; // each lane owns 2 channels
  float accA = 0.0f, accB = 0.0f, dacc = 0.0f;
  int rcnt = 0, cur = -1;

  auto flush = [&]() {
    if (cur < 0) return;
    unsafeAtomicAdd(&num[(n * C_ + cA) * CELLS_ + cur], accA);
    unsafeAtomicAdd(&num[(n * C_ + cB) * CELLS_ + cur], accB);
    float d = dacc;
#pragma unroll
    for (int o = 16; o > 0; o >>= 1) d += __shfl_xor(d, o, 32);
    if (lane == 0) {
      unsafeAtomicAdd(&den[n * CELLS_ + cur], d);
      unsafeAtomicAdd(&cnt[n * CELLS_ + cur], (float)(rcnt * 32));
    }
  };

  for (int t = 0; t < 4; ++t) {
    int w = bw + (t >> 1);
    int pbase = w * H_ + bh + ((t & 1) << 5);  // 32-px tile base
    int p  = pbase + lane;
    int gp = n * WH_ + p;

    float nrm = nrmC[gp];
    int  cell = celC[gp];
    float e   = __expf(nrm - m[n * CELLS_ + cell]);

    // ---- stage the 32x64 X tile into LDS (transposed, padded stride) ----
#if HAVE_ASYNC_B64
    wait_dscnt0();   // prior tile's DS reads done before async LDS writes land
    {
      const int c  = 2 * 0;  // pattern: instr i covers channels 2i, 2i+1
      (void)c;
#pragma unroll 8
      for (int i = 0; i < 32; ++i) {
        int ch = 2 * i + (lane >> 4);
        int px = 2 * (lane & 15);
        __builtin_amdgcn_global_load_async_to_lds_b64(
            (AS1 v2i_t*)(Xn + ch * WH_ + pbase + px),
            (AS3 v2i_t*)(ldsX + ch * XSTR + px), 0, 0);
      }
    }
#elif HAVE_ASYNC_LDS
    wait_dscnt0();
#pragma unroll 8
    for (int c = 0; c < C_; ++c)
      __builtin_amdgcn_global_load_async_to_lds_b32(
          (AS1 int*)(Xn + c * WH_ + p),
          (AS3 int*)(ldsX + c * XSTR + lane), 0, 0);
#else
#pragma unroll 8
    for (int c = 0; c < C_; ++c) ldsX[c * XSTR + lane] = Xn[c * WH_ + p];
#endif
    ldsE[lane] = e;
#if HAVE_ASYNC_LDS || HAVE_ASYNC_B64
    wait_asynccnt0();
#endif

    int c0 = __builtin_amdgcn_readfirstlane(cell);
    bool doacc = false;
    if (__all(cell == c0)) {
      if (c0 != cur) {
        flush();
        cur = c0;
        accA = accB = dacc = 0.0f; rcnt = 0;
      }
      doacc = true;
    } else {
      flush();
      cur = -1; accA = accB = dacc = 0.0f; rcnt = 0;
      // slow path: per-lane atomics (rare: mixed-cell tile)
      unsafeAtomicAdd(&den[n * CELLS_ + cell], e);
      unsafeAtomicAdd(&cnt[n * CELLS_ + cell], 1.0f);
#pragma unroll 8
      for (int c = 0; c < C_; ++c)
        unsafeAtomicAdd(&num[(n * C_ + c) * CELLS_ + cell],
                        e * ldsX[c * XSTR + lane]);
    }

    if (doacc) {
      dacc += e; rcnt += 1;
#pragma unroll 8
      for (int p2 = 0; p2 < 32; ++p2) {
        float ee = ldsE[p2];
        accA = fmaf(ee, ldsX[cA * XSTR + p2], accA);  // bank-conflict-free
        accB = fmaf(ee, ldsX[cB * XSTR + p2], accB);
      }
    }
  }
  flush();
}

// ---------------- kernel D: finalize out = num / den_full (in place) ----
__global__ void k_final(const float* __restrict__ ws, float* __restrict__ out) {
  int i = blockIdx.x * blockDim.x + threadIdx.x;
  if (i >= N_ * C_ * CELLS_) return;
  int cell = i & (CELLS_ - 1);
  int n    = i >> 16;                    // C_*CELLS_ == 65536
  int idx  = n * CELLS_ + cell;
  const float* m   = ws + M_OFF;
  const float* den = ws + DEN_OFF;
  const float* cnt = ws + CNT_OFF;
  float d = den[idx] + ((float)WH_ - cnt[idx]) * __expf(-m[idx]);
  out[i] = out[i] / d;
}

// ---------------- launcher ----------------
extern "C" void kernel_launch(void* const* d_in, const int* in_sizes, int n_in,
                              void* d_out, int out_size, void* d_ws, size_t ws_size,
                              hipStream_t stream) {
  const float* X  = (const float*)d_in[0];
  float* ws  = (float*)d_ws;
  float* out = (float*)d_out;   // doubles as the `num` accumulator

  // zero accumulators: m/den/cnt in ws, num in d_out
  int zn = NRM_OFF;  // 49152
  k_zero<<<(zn + 255) / 256, 256, 0, stream>>>(ws, zn);
  int on = N_ * C_ * CELLS_;  // 1048576
  k_zero<<<(on + 255) / 256, 256, 0, stream>>>(out, on);

  // 2048 waves -> 512 blocks of 128 threads (4 waves)
  k_norm   <<<512, 128, 0, stream>>>(X, ws);
  k_scatter<<<512, 128, 0, stream>>>(X, ws, out);
  k_final  <<<(on + 255) / 256, 256, 0, stream>>>(ws, out);
}